// Piezo_block_50276887166994
// MI455X (gfx1250) — compile-verified
//
#include <hip/hip_runtime.h>
#include <cmath>

typedef float v2f __attribute__((ext_vector_type(2)));
typedef float v8f __attribute__((ext_vector_type(8)));

// ---------------------------------------------------------------------------
// Setup kernel: compute e3nn Wigner-3j tensors in fp64 on device (exactly the
// reference's _cg/_q/einsum math), fold in constant_w / ALPHA / COB / IDX, and
// emit the transposed weight matrix Bt[32][64] (N padded 18->32) into d_ws.
// ---------------------------------------------------------------------------

struct cdbl { double re, im; };
__device__ inline cdbl cmul(cdbl a, cdbl b){ return {a.re*b.re - a.im*b.im, a.re*b.im + a.im*b.re}; }
__device__ inline cdbl cconj(cdbl a){ return {a.re, -a.im}; }
__device__ inline cdbl cadd(cdbl a, cdbl b){ return {a.re+b.re, a.im+b.im}; }
__device__ inline cdbl cscale(cdbl a, double s){ return {a.re*s, a.im*s}; }

__device__ inline int imini(int a, int b){ return a < b ? a : b; }
__device__ inline int imaxi(int a, int b){ return a > b ? a : b; }

__device__ double factd(int n){ double r = 1.0; for(int i = 2; i <= n; ++i) r *= (double)i; return r; }

__device__ double cgq(int j1,int m1,int j2,int m2,int j3,int m3){
  if (m3 != m1 + m2) return 0.0;
  int vmin = imaxi(imaxi(-j1 + j2 + m3, -j1 + m1), 0);
  int vmax = imini(imini(j2 + j3 + m1, j3 - j1 + j2), j3 + m3);
  double c = sqrt((double)(2*j3+1) * factd(j3+j1-j2) * factd(j3-j1+j2) * factd(j1+j2-j3)
                  * factd(j3+m3) * factd(j3-m3)
                  / (factd(j1+j2+j3+1) * factd(j1-m1) * factd(j1+m1) * factd(j2-m2) * factd(j2+m2)));
  double s = 0.0;
  for (int v = vmin; v <= vmax; ++v){
    double sg = ((v + j2 + m2) & 1) ? -1.0 : 1.0;
    s += sg * factd(j2+j3+m1-v) * factd(j1-m1+v)
         / (factd(v) * factd(j3-j1+j2-v) * factd(j3+m3-v) * factd(v+j1-j2-m3));
  }
  return c * s;
}

// real->complex SH change of basis, e3nn convention, q is (2l+1)^2 row-major
__device__ void qmat(int l, cdbl* q){
  int n = 2*l + 1;
  for (int t = 0; t < n*n; ++t) q[t] = {0.0, 0.0};
  const double is2 = 0.70710678118654752440;
  for (int m = -l; m < 0; ++m){
    q[(l+m)*n + (l-m)] = { is2, 0.0 };
    q[(l+m)*n + (l+m)] = { 0.0, -is2 };
  }
  q[l*n + l] = {1.0, 0.0};
  for (int m = 1; m <= l; ++m){
    double sg = (m & 1) ? -1.0 : 1.0;
    q[(l+m)*n + (l+m)] = { sg*is2, 0.0 };
    q[(l+m)*n + (l-m)] = { 0.0, sg*is2 };
  }
  cdbl ph = {1.0, 0.0}; const cdbl mi = {0.0, -1.0};
  for (int t = 0; t < l; ++t) ph = cmul(ph, mi);
  for (int t = 0; t < n*n; ++t) q[t] = cmul(q[t], ph);
}

__device__ double w3j_elem(int l1,int l2,int l3,
                           const cdbl* q1,const cdbl* q2,const cdbl* q3,
                           int a,int b,int c){
  int n1 = 2*l1+1, n2 = 2*l2+1, n3 = 2*l3+1;
  cdbl s = {0.0, 0.0};
  for (int i = 0; i < n1; ++i)
    for (int k = 0; k < n2; ++k){
      int m = (i - l1) + (k - l2) + l3;       // m3 = m1 + m2
      if (m < 0 || m >= n3) continue;
      double cv = cgq(l1, i-l1, l2, k-l2, l3, m-l3);
      if (cv == 0.0) continue;
      cdbl t = cmul(cmul(q1[i*n1 + a], q2[k*n2 + b]), cconj(q3[m*n3 + c]));
      s = cadd(s, cscale(t, cv));
    }
  return s.re;
}

__device__ __constant__ int IDXc[6] = {0, 4, 8, 1, 5, 6};

__global__ __launch_bounds__(256) void piezo_setup_b(const float* __restrict__ w,
                                                     float* __restrict__ Bt){
  __shared__ cdbl Q0[1], Q1[9], Q2[25], Q3[49];
  __shared__ double W101s[9], W121s[45], W221s[75], W321s[105], W110s[9], W112s[45];
  __shared__ double COBs[6][9];
  const int tid = threadIdx.x;

  if (tid == 0){ qmat(0, Q0); qmat(1, Q1); qmat(2, Q2); qmat(3, Q3); }
  __syncthreads();

  // 288 tensor elements total, one per thread (with a wrap for the overflow)
  for (int idx = tid; idx < 288; idx += blockDim.x){
    if (idx < 9){                                  // W3J(1,0,1): (3,1,3)
      int a = idx/3, c = idx%3;
      W101s[idx] = w3j_elem(1,0,1, Q1,Q0,Q1, a,0,c);
    } else if (idx < 54){                          // W3J(1,2,1): (3,5,3)
      int e = idx - 9; int a = e/15, b = (e/3)%5, c = e%3;
      W121s[e] = w3j_elem(1,2,1, Q1,Q2,Q1, a,b,c);
    } else if (idx < 129){                         // W3J(2,2,1): (5,5,3)
      int e = idx - 54; int a = e/15, b = (e/3)%5, c = e%3;
      W221s[e] = w3j_elem(2,2,1, Q2,Q2,Q1, a,b,c);
    } else if (idx < 234){                         // W3J(3,2,1): (7,5,3)
      int e = idx - 129; int a = e/15, b = (e/3)%5, c = e%3;
      W321s[e] = w3j_elem(3,2,1, Q3,Q2,Q1, a,b,c);
    } else if (idx < 243){                         // W3J(1,1,0): (3,3,1)
      int e = idx - 234; int a = e/3, b = e%3;
      W110s[e] = w3j_elem(1,1,0, Q1,Q1,Q0, a,b,0);
    } else {                                       // W3J(1,1,2): (3,3,5)
      int e = idx - 243; int a = e/15, b = (e/5)%3, c = e%5;
      W112s[e] = w3j_elem(1,1,2, Q1,Q1,Q2, a,b,c);
    }
  }
  __syncthreads();

  // Frobenius-normalize each tensor (one thread per tensor; tiny work)
  if (tid < 6){
    double* arr; int n;
    switch (tid){
      case 0: arr = W101s; n = 9;   break;
      case 1: arr = W121s; n = 45;  break;
      case 2: arr = W221s; n = 75;  break;
      case 3: arr = W321s; n = 105; break;
      case 4: arr = W110s; n = 9;   break;
      default: arr = W112s; n = 45; break;
    }
    double nrm = 0.0;
    for (int t = 0; t < n; ++t) nrm += arr[t]*arr[t];
    nrm = sqrt(nrm);
    for (int t = 0; t < n; ++t) arr[t] /= nrm;
  }
  __syncthreads();

  // COB (6,3,3): row0 = W110[:,:,0]; rows 1..5 = sqrt(5)*moveaxis(W112,2,0)
  if (tid == 0){
    const double s5 = sqrt(5.0);
    for (int p = 0; p < 3; ++p)
      for (int q = 0; q < 3; ++q){
        COBs[0][3*p+q] = W110s[p*3 + q];               // (p*3+q)*1 + 0
        for (int m = 0; m < 5; ++m)
          COBs[1+m][3*p+q] = s5 * W112s[(p*3+q)*5 + m];
      }
  }
  __syncthreads();

  // Bt[n][k] = sum over channels: weight so that out[b][n] = sum_k X[b][k]*Bt[n][k]
  const double ALPHA = 0.61237243569579452455;  // sqrt(3/8)
  for (int e = tid; e < 32*64; e += blockDim.x){
    int n = e >> 6, k = e & 63;
    float val = 0.0f;
    if (n < 18){
      int kOut = n / 6, t = n % 6;
      int pq = IDXc[t];
      double v = 0.0;
      if (k >= 10 && k < 16){                 // x1o: paths w[0:2] (->0e) and w[2:4] (->2e)
        int u = (k-10)/3, i = (k-10)%3;
        v += (double)w[u] * W101s[i*3 + kOut] * COBs[0][pq];
        for (int j = 0; j < 5; ++j)
          v += (double)w[2+u] * W121s[(i*5+j)*3 + kOut] * COBs[1+j][pq];
      } else if (k >= 26 && k < 36){          // x2o: paths w[4:6]
        int u = (k-26)/5, i = (k-26)%5;
        for (int j = 0; j < 5; ++j)
          v += (double)w[4+u] * W221s[(i*5+j)*3 + kOut] * COBs[1+j][pq];
      } else if (k >= 50){                    // x3o: paths w[6:8]
        int u = (k-50)/7, i = (k-50)%7;
        for (int j = 0; j < 5; ++j)
          v += (double)w[6+u] * W321s[(i*5+j)*3 + kOut] * COBs[1+j][pq];
      }
      val = (float)(ALPHA * v);
    }
    Bt[n*64 + k] = val;
  }
}

// ---------------------------------------------------------------------------
// Main kernel: out(1e6 x 18) = X(1e6 x 64) * B(64 x 18) via v_wmma_f32_16x16x4_f32.
// Each wave: preload B fragments (held in VGPRs), then loop over 16-row tiles.
// ---------------------------------------------------------------------------

#define WPB 4   // waves per block
#define TPW 5   // 16-row tiles per wave

__global__ __launch_bounds__(WPB*32) void piezo_wmma_main(const float* __restrict__ X,
                                                          const float* __restrict__ Bt,
                                                          float* __restrict__ out,
                                                          int rows){
  const int lane = threadIdx.x & 31;
  const int wave = threadIdx.x >> 5;
  const int half = lane >> 4;     // 0: lanes 0-15, 1: lanes 16-31
  const int l16  = lane & 15;

  // B fragments: A/B fp32 16x16x4 layout -> lane holds (K = 4*kk + 2*half + {0,1})
  v2f bf0[16], bf1[16];
  #pragma unroll
  for (int kk = 0; kk < 16; ++kk){
    int k0 = 4*kk + 2*half;
    bf0[kk] = *(const v2f*)(Bt + (l16     ) * 64 + k0);  // N-tile 0: cols 0..15
    bf1[kk] = *(const v2f*)(Bt + (l16 + 16) * 64 + k0);  // N-tile 1: cols 16..31 (18..31 zero)
  }

  const long tilebase = (long)(blockIdx.x * WPB + wave) * TPW;
  for (int t = 0; t < TPW; ++t){
    long row0 = (tilebase + t) * 16;
    if (row0 + 16 > (long)rows) break;               // wave-uniform: EXEC stays all-ones
    const float* xr = X + (row0 + l16) * 64 + 2*half;

    // prefetch next tile's rows (speculative; dropped if OOB)
    __builtin_prefetch((const void*)(xr + 16*64), 0, 1);
    __builtin_prefetch((const void*)(xr + 16*64 + 32), 0, 1);

    v8f acc0 = {}; v8f acc1 = {};
    #pragma unroll
    for (int kk = 0; kk < 16; ++kk){
      v2f a = *(const v2f*)(xr + 4*kk);
      acc0 = __builtin_amdgcn_wmma_f32_16x16x4_f32(false, a, false, bf0[kk],
                                                   (short)0, acc0, false, false);
      acc1 = __builtin_amdgcn_wmma_f32_16x16x4_f32(false, a, false, bf1[kk],
                                                   (short)0, acc1, false, false);
    }

    // D layout: VGPR r, lanes 0-15 -> row r, lanes 16-31 -> row r+8; col = lane&15
    float* orow = out + row0 * 18;
    #pragma unroll
    for (int r = 0; r < 8; ++r){
      int row = r + half*8;
      orow[row*18 + l16] = acc0[r];
      if (l16 < 2) orow[row*18 + 16 + l16] = acc1[r];
    }
  }
}

// ---------------------------------------------------------------------------

extern "C" void kernel_launch(void* const* d_in, const int* in_sizes, int n_in,
                              void* d_out, int out_size, void* d_ws, size_t ws_size,
                              hipStream_t stream){
  const float* X = (const float*)d_in[0];   // (BS, 64) fp32
  const float* w = (const float*)d_in[1];   // (8,) fp32
  float* out = (float*)d_out;               // (BS, 3, 6) fp32
  float* Bt  = (float*)d_ws;                // 32*64 floats = 8 KB

  const int rows  = in_sizes[0] / 64;
  const int tiles = rows / 16;              // BS = 1e6 -> 62500 tiles exactly
  const int blocks = (tiles + WPB*TPW - 1) / (WPB*TPW);

  piezo_setup_b<<<1, 256, 0, stream>>>(w, Bt);
  piezo_wmma_main<<<blocks, WPB*32, 0, stream>>>(X, Bt, out, rows);
}